// PatchTransformer_28948079575168
// MI455X (gfx1250) — compile-verified
//
#include <hip/hip_runtime.h>
#include <math.h>

typedef __attribute__((ext_vector_type(2))) float v2f;
typedef __attribute__((ext_vector_type(8))) float v8f;

#define IMG_H 416
#define IMG_W 416
#define NIMG  128
#define TILES_X 26
#define TILES_Y 26
#define NTILES (TILES_X * TILES_Y)   // 676

// ---------------------------------------------------------------------------
// Kernel 1: per-image affine theta (128 tiny transforms) -> workspace
// ---------------------------------------------------------------------------
__global__ void theta_kernel(const float* __restrict__ bboxes,
                             const float* __restrict__ angle,
                             const int*   __restrict__ patch_ori_size,
                             float* __restrict__ theta) {
    int n = blockIdx.x * blockDim.x + threadIdx.x;
    if (n >= NIMG) return;
    const float* bb = bboxes + 6 * n;
    float x1 = bb[0], y1 = bb[1], x2 = bb[2], y2 = bb[3];
    float bw = (x2 - x1) * (float)IMG_W;
    float bh = (y2 - y1) * (float)IMG_H;
    float cx = (x1 + x2) * 0.5f;
    float cy = (y1 + y2) * 0.5f - (y2 - y1) * 0.1f;
    float tx = (0.5f - cx) * 2.0f;
    float ty = (0.5f - cy) * 2.0f;
    float target = 0.2f * sqrtf(bw * bw + bh * bh);
    float inv_scale = (float)patch_ori_size[0] / target;   // 1/scale
    float s = sinf(angle[n]);
    float c = cosf(angle[n]);
    float* t = theta + 6 * n;
    t[0] = c * inv_scale;                      // t00
    t[1] = s * inv_scale;                      // t01
    t[2] = (tx * c + ty * s) * inv_scale;      // t02
    t[3] = -s * inv_scale;                     // t10
    t[4] = c * inv_scale;                      // t11
    t[5] = (-tx * s + ty * c) * inv_scale;     // t12
}

// ---------------------------------------------------------------------------
// Kernel 2: one wave32 per 16x16 output tile.
// Affine grid via V_WMMA_F32_16X16X4_F32 (grid = P x theta^T), then guarded
// bilinear gather + non-temporal stores (output is write-once, > L2).
// ---------------------------------------------------------------------------
__global__ __launch_bounds__(256)
void sample_kernel(const float* __restrict__ img,
                   const float* __restrict__ theta,
                   float* __restrict__ out) {
    const int n    = blockIdx.y;                 // image index
    const int lane = threadIdx.x & 31;
    const int wave = threadIdx.x >> 5;
    int tile = blockIdx.x * 8 + wave;
    tile = (tile < NTILES) ? tile : (NTILES - 1);   // idempotent duplicate work
    const int x0 = (tile % TILES_X) * 16;
    const int y0 = (tile / TILES_X) * 16;

    const float* th = theta + 6 * n;
    const float t00 = th[0], t01 = th[1], t02 = th[2];
    const float t10 = th[3], t11 = th[4], t12 = th[5];

    const int hi  = lane >> 4;     // K-half (A/B) and M-half (D)
    const int sub = lane & 15;     // A: row M; B/D: col N

    // normalized coords of this lane's row (for A) and column (for B)
    const float ysm = (2.0f * (float)(y0 + sub) + 1.0f) * (1.0f / (float)IMG_H) - 1.0f;
    const float xsn = (2.0f * (float)(x0 + sub) + 1.0f) * (1.0f / (float)IMG_W) - 1.0f;

    // A(m,k) = [ys(m), 1, 0, 0]; lanes>=16 carry K=2,3 (zero)
    v2f A, Bx, By;
    A[0]  = hi ? 0.0f : ysm;
    A[1]  = hi ? 0.0f : 1.0f;
    // gx: B(0,n)=t01, B(1,n)=t00*xs(n)+t02
    Bx[0] = hi ? 0.0f : t01;
    Bx[1] = hi ? 0.0f : fmaf(t00, xsn, t02);
    // gy: B(0,n)=t11, B(1,n)=t10*xs(n)+t12
    By[0] = hi ? 0.0f : t11;
    By[1] = hi ? 0.0f : fmaf(t10, xsn, t12);

    v8f zero = {};
    // D(m,n) = gx / gy over the 16x16 tile (f32 precision, exact affine grid)
    v8f gx = __builtin_amdgcn_wmma_f32_16x16x4_f32(false, A, false, Bx,
                                                   (short)0, zero, false, false);
    v8f gy = __builtin_amdgcn_wmma_f32_16x16x4_f32(false, A, false, By,
                                                   (short)0, zero, false, false);

    const size_t HW = (size_t)IMG_H * IMG_W;
    const float* imgn = img + (size_t)n * 3 * HW;
    float*       outn = out + (size_t)n * 3 * HW;
    const int xp = x0 + sub;                 // this lane's output column

#pragma unroll
    for (int i = 0; i < 8; ++i) {
        const int yp = y0 + 8 * hi + i;      // D layout: M = i + 8*(lane>=16)
        const float fx = gx[i];
        const float fy = gy[i];
        // ix = ((gx+1)*W - 1)*0.5
        const float ixf = fmaf(fx + 1.0f, 208.0f, -0.5f);
        const float iyf = fmaf(fy + 1.0f, 208.0f, -0.5f);
        const float xf0 = floorf(ixf);
        const float yf0 = floorf(iyf);
        const float wx1 = ixf - xf0, wy1 = iyf - yf0;
        const float wx0 = 1.0f - wx1, wy0 = 1.0f - wy1;

        float r = 0.0f, g = 0.0f, b = 0.0f;
        // any of the 4 taps in-bounds?  (true for ~1% of pixels, wave-coherent)
        if (xf0 >= -1.0f && xf0 < 416.0f && yf0 >= -1.0f && yf0 < 416.0f) {
            const int xi0 = (int)xf0, yi0 = (int)yf0;
            const int xi1 = xi0 + 1,  yi1 = yi0 + 1;
            const bool vx0 = (xi0 >= 0) & (xi0 < IMG_W);
            const bool vx1 = (xi1 >= 0) & (xi1 < IMG_W);
            if ((yi0 >= 0) & (yi0 < IMG_H)) {
                const float* row = imgn + (size_t)yi0 * IMG_W;
                if (vx0) { float w = wx0 * wy0;
                    r = fmaf(row[xi0],          w, r);
                    g = fmaf(row[HW + xi0],     w, g);
                    b = fmaf(row[2 * HW + xi0], w, b); }
                if (vx1) { float w = wx1 * wy0;
                    r = fmaf(row[xi1],          w, r);
                    g = fmaf(row[HW + xi1],     w, g);
                    b = fmaf(row[2 * HW + xi1], w, b); }
            }
            if ((yi1 >= 0) & (yi1 < IMG_H)) {
                const float* row = imgn + (size_t)yi1 * IMG_W;
                if (vx0) { float w = wx0 * wy1;
                    r = fmaf(row[xi0],          w, r);
                    g = fmaf(row[HW + xi0],     w, g);
                    b = fmaf(row[2 * HW + xi0], w, b); }
                if (vx1) { float w = wx1 * wy1;
                    r = fmaf(row[xi1],          w, r);
                    g = fmaf(row[HW + xi1],     w, g);
                    b = fmaf(row[2 * HW + xi1], w, b); }
            }
        }
        // streaming output: NT stores (output > 192MB L2, never re-read)
        const size_t o = (size_t)yp * IMG_W + xp;
        __builtin_nontemporal_store(r, outn + o);
        __builtin_nontemporal_store(g, outn + HW + o);
        __builtin_nontemporal_store(b, outn + 2 * HW + o);
    }
}

extern "C" void kernel_launch(void* const* d_in, const int* in_sizes, int n_in,
                              void* d_out, int out_size, void* d_ws, size_t ws_size,
                              hipStream_t stream) {
    (void)in_sizes; (void)n_in; (void)out_size; (void)ws_size;
    const float* adv    = (const float*)d_in[0];   // (8,16,3,416,416) f32
    const float* bboxes = (const float*)d_in[1];   // (8,16,6) f32
    const float* angle  = (const float*)d_in[2];   // (128,) f32
    const int*   pos    = (const int*)d_in[3];     // scalar 300
    float* out   = (float*)d_out;
    float* theta = (float*)d_ws;                   // 128*6 floats

    theta_kernel<<<1, 128, 0, stream>>>(bboxes, angle, pos, theta);

    dim3 grid((NTILES + 7) / 8, NIMG);             // (85, 128), 8 waves/block
    sample_kernel<<<grid, 256, 0, stream>>>(adv, theta, out);
}